// HKS_2319282340281
// MI455X (gfx1250) — compile-verified
//
#include <hip/hip_runtime.h>

typedef __attribute__((ext_vector_type(2))) float v2f;
typedef __attribute__((ext_vector_type(8))) float v8f;

#define KDIM   128
#define COUNT  16
#define WAVES  4           // waves per block
#define ROWS_PER_WAVE 16
#define ROWS_PER_BLOCK (WAVES * ROWS_PER_WAVE)   // 64

// out[b,v,c] = sum_k evecs[b,v,k]^2 * exp(-evals[b,k] * scale[c])
// scale[c] = 10^(-2 + 2c/(COUNT-1))
__global__ __launch_bounds__(WAVES * 32) void hks_wmma_kernel(
    const float* __restrict__ evals,   // [B][KDIM]
    const float* __restrict__ evecs,   // [B][V][KDIM]
    float* __restrict__ out,           // [B][V][COUNT]
    int V)
{
    // coefsPair[p][c] = { coefsT[2p][c], coefsT[2p+1][c] } where
    // coefsT[k][c] = exp(-evals[b,k]*scale[c])   (B-matrix, K-major pairs)
    __shared__ float2 coefsPair[KDIM / 2][COUNT];           // 8 KB
    __shared__ float  tileA[WAVES][ROWS_PER_WAVE][KDIM];    // 32 KB

    const int b    = blockIdx.y;
    const int tid  = threadIdx.x;
    const int lane = tid & 31;
    const int wave = tid >> 5;

    // ---- build coefs for this batch: 16 x 128 values, 128 threads x 16 ----
    {
        const int c     = tid >> 3;          // 0..15
        const int kbase = (tid & 7) * 16;    // 0,16,...,112
        const float scale = exp10f(-2.0f + (float)c * (2.0f / (float)(COUNT - 1)));
        #pragma unroll
        for (int kk = 0; kk < 16; ++kk) {
            const int k = kbase + kk;
            const float e = evals[b * KDIM + k];
            ((float*)&coefsPair[k >> 1][c])[k & 1] = __expf(-e * scale);
        }
    }
    __syncthreads();   // all 128 threads reach this (pre-divergence)

    const int row0 = blockIdx.x * ROWS_PER_BLOCK + wave * ROWS_PER_WAVE;
    if (row0 < V) {    // V % 16 == 0 -> full tiles only; uniform per wave
        // ---- stage 16x128 evecs tile into this wave's LDS slab ----
        // iteration j loads row j: 32 lanes x float4 = full 512B row, coalesced
        const float* __restrict__ gsrc = evecs + ((size_t)b * V + row0) * KDIM;
        float* __restrict__ ldst = &tileA[wave][0][0];
        #pragma unroll
        for (int j = 0; j < ROWS_PER_WAVE; ++j) {
            const float4 t = *(const float4*)(gsrc + j * KDIM + (lane << 2));
            *(float4*)(ldst + j * KDIM + (lane << 2)) = t;
        }
        // wave-private slab: only need this wave's DS stores complete
        asm volatile("s_wait_dscnt 0x0" ::: "memory");

        // ---- 32 x V_WMMA_F32_16X16X4_F32, K marches 4 per step ----
        // A frag (16x4 f32): lanes 0-15 -> M=lane, K = k0+{0,1}
        //                    lanes 16-31 -> M=lane-16, K = k0+{2,3}
        // B frag (4x16 f32): VGPR pair = K pair, lane half selects K+2
        const int half = lane >> 4;       // 0 or 1
        const int mr   = lane & 15;
        const float2* __restrict__ arow = (const float2*)(&tileA[wave][mr][0]);

        v8f acc = {0.f, 0.f, 0.f, 0.f, 0.f, 0.f, 0.f, 0.f};
        #pragma unroll
        for (int t = 0; t < KDIM / 4; ++t) {
            const int p = (t << 1) + half;        // float2 pair index = k0/2 + half
            const float2 av = arow[p];            // raw evecs pair
            const float2 bv = coefsPair[p][mr];   // coefs pair
            v2f a; a[0] = av.x * av.x; a[1] = av.y * av.y;   // square at use
            v2f bb; bb[0] = bv.x; bb[1] = bv.y;
            acc = __builtin_amdgcn_wmma_f32_16x16x4_f32(
                false, a, false, bb, (short)0, acc, false, false);
        }

        // ---- store: VGPR r holds row (row0 + half*8 + r), column mr ----
        float* __restrict__ orow = out + ((size_t)b * V + row0 + (half << 3)) * COUNT + mr;
        #pragma unroll
        for (int r = 0; r < 8; ++r)
            orow[r * COUNT] = acc[r];
    }
}

extern "C" void kernel_launch(void* const* d_in, const int* in_sizes, int n_in,
                              void* d_out, int out_size, void* d_ws, size_t ws_size,
                              hipStream_t stream) {
    const float* evals = (const float*)d_in[0];   // [B][128] f32
    const float* evecs = (const float*)d_in[1];   // [B][V][128] f32
    // d_in[2] is `count` (==16), baked into COUNT

    const int B = in_sizes[0] / KDIM;
    const int V = in_sizes[1] / (B * KDIM);

    dim3 grid((V + ROWS_PER_BLOCK - 1) / ROWS_PER_BLOCK, B);
    hks_wmma_kernel<<<grid, WAVES * 32, 0, stream>>>(evals, evecs, (float*)d_out, V);
}